// ZGate_549755814195
// MI455X (gfx1250) — compile-verified
//
#include <hip/hip_runtime.h>
#include <stdint.h>

// Z-gate (D=4, S=1, wires {0,2,5,8}, L=10) on a batch of B=16 state columns.
// U is diagonal with entries i^t, t = (sum of 4 base-4 digits of n) mod 4.
// Pure streaming op: 268 MB traffic -> ~11.5 us floor at 23.3 TB/s.
// Data movement uses the CDNA5 async global->LDS path (ASYNCcnt-tracked);
// output uses non-temporal b128 stores (write-once data, keep L2 for input).

#define DEV_INLINE __device__ __forceinline__

typedef int   v4i __attribute__((vector_size(16)));     // matches builtin pointee type
typedef float v4f __attribute__((ext_vector_type(4)));  // native vector (NT-store legal)

DEV_INLINE void async_load_b128_to_lds(const float* gp, float* lp) {
  // Low 32 bits of a generic pointer into __shared__ memory == LDS byte offset
  const uint32_t lofs = (uint32_t)(uintptr_t)lp;
#if __has_builtin(__builtin_amdgcn_global_load_async_to_lds_b128)
  __builtin_amdgcn_global_load_async_to_lds_b128(
      (__attribute__((address_space(1))) v4i*)(uintptr_t)gp,
      (__attribute__((address_space(3))) v4i*)lofs,
      /*imm offset=*/0, /*cpol=*/0);
#else
  asm volatile("global_load_async_to_lds_b128 %0, %1, off"
               :
               : "v"(lofs), "v"(gp)
               : "memory");
#endif
}

DEV_INLINE void wait_asynccnt_zero() {
#if __has_builtin(__builtin_amdgcn_s_wait_asynccnt)
  __builtin_amdgcn_s_wait_asynccnt(0);
#else
  asm volatile("s_wait_asynccnt 0" ::: "memory");
#endif
}

// x:   [2, N, B] f32 (real plane then imag plane), nb = N*B
// out: complex64 [N, B] viewed as interleaved f32 pairs (re, im)
__global__ __launch_bounds__(256) void zgate_phase_kernel(
    const float* __restrict__ x, float* __restrict__ out, unsigned nb) {
  __shared__ __attribute__((aligned(16))) float lds_re[1024];
  __shared__ __attribute__((aligned(16))) float lds_im[1024];

  const unsigned tid  = threadIdx.x;
  const unsigned base = blockIdx.x * 1024u;   // 1024 complex elems per block
  if (base >= nb) return;                     // uniform per block (nb % 1024 == 0)

  // 4 consecutive complex elements per thread; e0 % 4 == 0 and B == 16
  // => all 4 share the same basis index n = e0 >> 4.
  const unsigned e0 = base + tid * 4u;

  const float* Rp = x;                  // real plane
  const float* Ip = x + (size_t)nb;     // imag plane

  // Async DMA 16B of real and 16B of imag into LDS (2 outstanding / thread).
  // No barrier needed: each thread reads back only its own LDS slots, and
  // ASYNCcnt==0 guarantees this wave's async fills have landed.
  async_load_b128_to_lds(Rp + e0, &lds_re[tid * 4u]);
  async_load_b128_to_lds(Ip + e0, &lds_im[tid * 4u]);
  wait_asynccnt_zero();

  const v4f re = *reinterpret_cast<const v4f*>(&lds_re[tid * 4u]);
  const v4f im = *reinterpret_cast<const v4f*>(&lds_im[tid * 4u]);

  // t = (d0 + d2 + d5 + d8) mod 4; digit i lives at bit 2*(9-i).
  const unsigned n = e0 >> 4;
  const unsigned t = ((n >> 18) + (n >> 14) + (n >> 8) + (n >> 2)) & 3u;
  // phase i^t = c + i*s with (c,s) in {(1,0),(0,1),(-1,0),(0,-1)}
  const float c = (t == 0u) ? 1.0f : ((t == 2u) ? -1.0f : 0.0f);
  const float s = (t == 1u) ? 1.0f : ((t == 3u) ? -1.0f : 0.0f);

  v4f o0, o1;  // interleaved (re, im) pairs
  o0.x = c * re.x - s * im.x;  o0.y = s * re.x + c * im.x;
  o0.z = c * re.y - s * im.y;  o0.w = s * re.y + c * im.y;
  o1.x = c * re.z - s * im.z;  o1.y = s * re.z + c * im.z;
  o1.z = c * re.w - s * im.w;  o1.w = s * re.w + c * im.w;

  // Write-once output: stream past the caches with non-temporal b128 stores.
  v4f* op = reinterpret_cast<v4f*>(out + 2ull * (size_t)e0);  // 16B-aligned
  __builtin_nontemporal_store(o0, op);
  __builtin_nontemporal_store(o1, op + 1);
}

extern "C" void kernel_launch(void* const* d_in, const int* in_sizes, int n_in,
                              void* d_out, int out_size, void* d_ws, size_t ws_size,
                              hipStream_t stream) {
  (void)n_in; (void)out_size; (void)d_ws; (void)ws_size;
  const float* x  = (const float*)d_in[0];
  float* out      = (float*)d_out;
  const unsigned nb = (unsigned)(in_sizes[0] / 2);   // N*B complex elements
  const unsigned blocks = (nb + 1023u) / 1024u;      // exact for N*B = 2^24
  zgate_phase_kernel<<<dim3(blocks), dim3(256), 0, stream>>>(x, out, nb);
}